// ManeuverHead_43971875176951
// MI455X (gfx1250) — compile-verified
//
#include <hip/hip_runtime.h>

// ManeuverHead fused kernels for gfx1250 (MI455X).
// Inputs (setup_inputs order):
//  0 node_features   f32 [N,128]
//  1 global_features f32 [B,128]
//  2 group_mask      bool -> int32 [N]
//  3 maneuver_valid_mask bool -> int32 [B,56]
//  4 batch           int32 [N] (sorted)
//  5 w1 f32 [256,128]   6 b1 f32 [128]
//  7 w2 f32 [128,7]     8 b2 f32 [7]
// Output: f32 [B, 56]

typedef float v2f __attribute__((ext_vector_type(2)));
typedef float v8f __attribute__((ext_vector_type(8)));

#define DIM   128
#define GMAX  8
#define RDIR  7
#define NEGV  (-1e9f)
#define HSTRIDE 130   // 128 + 2 pad floats -> conflict-free column reads

// ---- workspace layout (bytes) ----
// [0,               B*8*4)        idxbuf  (int)
// [B*8*4,           B*8*4+512)    zero row (float[128])
// [.. +512,         +512+131072)  w1 packed fragments (64 ksteps * 8 ntiles * 32 lanes * 2 f32)
// [.. ,             +8192)        w2 packed fragments (32 ksteps * 32 lanes * 2 f32)
#define W1P_FRAGS (64 * 8 * 32)
#define W2P_FRAGS (32 * 32)

__device__ __forceinline__ int lower_bound_i32(const int* __restrict__ a, int n, int key) {
  int lo = 0, hi = n;
  while (lo < hi) {
    int mid = (lo + hi) >> 1;
    if (a[mid] < key) lo = mid + 1; else hi = mid;
  }
  return lo;
}

// Repack w1/w2 into WMMA B-fragment-major layout + build the zero row.
// B-frag layout for V_WMMA_F32_16X16X4_F32: VGPR0 = rows {K, K+2} (low/high lanes),
// VGPR1 = rows {K+1, K+3}; column = lane&15. Each lane's 2 floats stored contiguously
// so the hot loop does one coalesced global_load_b64 per fragment.
__global__ __launch_bounds__(256)
void mh_pack_kernel(const float* __restrict__ w1, const float* __restrict__ w2,
                    float* __restrict__ w1p, float* __restrict__ w2p,
                    float* __restrict__ zrow) {
  const int t = blockIdx.x * 256 + threadIdx.x;
  if (t < W1P_FRAGS) {
    const int lane = t & 31;
    const int n    = (t >> 5) & 7;
    const int k    = t >> 8;
    const int col  = k * 4 + ((lane >> 4) << 1);
    const int c    = n * 16 + (lane & 15);
    v2f f;
    f.x = w1[(col    ) * DIM + c];
    f.y = w1[(col + 1) * DIM + c];
    *(v2f*)(w1p + t * 2) = f;
  } else if (t < W1P_FRAGS + W2P_FRAGS) {
    const int u    = t - W1P_FRAGS;
    const int lane = u & 31;
    const int k    = u >> 5;
    const int col  = k * 4 + ((lane >> 4) << 1);
    const int n2   = lane & 15;
    v2f f;
    f.x = (n2 < RDIR) ? w2[(col    ) * RDIR + n2] : 0.f;
    f.y = (n2 < RDIR) ? w2[(col + 1) * RDIR + n2] : 0.f;
    *(v2f*)(w2p + u * 2) = f;
  } else if (t < W1P_FRAGS + W2P_FRAGS + DIM) {
    zrow[t - W1P_FRAGS - W2P_FRAGS] = 0.f;
  }
}

// One wave32 per graph: binary search sorted batch[] for the node range, then
// ballot-scan for the first <=8 group nodes.
__global__ __launch_bounds__(32)
void mh_slots_kernel(const int* __restrict__ batch, const int* __restrict__ gmask,
                     int* __restrict__ idxbuf, int N) {
  const int b    = blockIdx.x;
  const int lane = threadIdx.x;
  const int start = lower_bound_i32(batch, N, b);
  const int end   = lower_bound_i32(batch, N, b + 1);
  int cnt = 0;
  for (int base = start; base < end && cnt < GMAX; base += 32) {
    const int  i   = base + lane;
    const bool act = (i < end) && (gmask[i] != 0);
    const unsigned m = __builtin_amdgcn_ballot_w32(act);
    if (act) {
      const int r = cnt + __popc(m & ((1u << lane) - 1u));
      if (r < GMAX) idxbuf[b * GMAX + r] = i;
    }
    cnt += __popc(m);
  }
  if (cnt > GMAX) cnt = GMAX;
  if (lane >= cnt && lane < GMAX) idxbuf[b * GMAX + lane] = -1;
}

// One wave32 per 16-row tile (16 slots = 2 graphs). Fused gather + MLP + mask.
__global__ __launch_bounds__(32)
void mh_mlp_kernel(const float* __restrict__ nf, const float* __restrict__ gf,
                   const int* __restrict__ idxbuf, const int* __restrict__ mvmask,
                   const float* __restrict__ w1p, const float* __restrict__ b1,
                   const float* __restrict__ w2p, const float* __restrict__ b2,
                   const float* __restrict__ zrow, float* __restrict__ out) {
  __shared__ float hlds[16 * HSTRIDE];

  const int tile  = blockIdx.x;
  const int lane  = threadIdx.x;
  const int m     = lane & 15;        // A: row M; B/C/D: column
  const int khalf = (lane >> 4) << 1; // 0 for lanes 0-15, 2 for lanes 16-31
  const int row   = tile * 16 + m;
  const int idx   = idxbuf[row];      // gathered node index, -1 = empty
  const bool hasNode = (idx >= 0);

  // Empty slots read the zero row -> unconditional A loads, no EXEC churn.
  const float* __restrict__ aN = hasNode ? (nf + (long)idx * DIM) : zrow;
  const float* __restrict__ aG = gf + (long)(row >> 3) * DIM;

  // ---- Layer 1: acc[n] (16x16 f32) over K=256, N-tiles n=0..7 ----
  v8f acc[8];
#pragma unroll
  for (int n = 0; n < 8; n++) {
    const float bb = b1[n * 16 + m];  // whole accumulator column = n*16+m
#pragma unroll
    for (int j = 0; j < 8; j++) acc[n][j] = bb;
  }

  // K = 0..127 : gathered node features
#pragma unroll 2
  for (int k = 0; k < 32; k++) {
    const v2f a = *(const v2f*)(aN + k * 4 + khalf);
    const float* __restrict__ bp = w1p + ((k * 8) * 32 + lane) * 2;
#pragma unroll
    for (int n = 0; n < 8; n++) {
      const v2f bf = *(const v2f*)(bp + n * 64);
      acc[n] = __builtin_amdgcn_wmma_f32_16x16x4_f32(
          false, a, false, bf, (short)0, acc[n], false, false);
    }
  }
  // K = 128..255 : broadcast global features
#pragma unroll 2
  for (int k = 0; k < 32; k++) {
    const v2f a = *(const v2f*)(aG + k * 4 + khalf);
    const float* __restrict__ bp = w1p + (((k + 32) * 8) * 32 + lane) * 2;
#pragma unroll
    for (int n = 0; n < 8; n++) {
      const v2f bf = *(const v2f*)(bp + n * 64);
      acc[n] = __builtin_amdgcn_wmma_f32_16x16x4_f32(
          false, a, false, bf, (short)0, acc[n], false, false);
    }
  }

  // ---- ReLU + stage h[16,128] to LDS (D-layout -> row-major, padded stride) ----
  const int mrow = (lane >> 4) * 8;
#pragma unroll
  for (int n = 0; n < 8; n++) {
#pragma unroll
    for (int j = 0; j < 8; j++) {
      hlds[(j + mrow) * HSTRIDE + n * 16 + m] = fmaxf(acc[n][j], 0.f);
    }
  }
  __syncthreads();

  // ---- Layer 2: h[16,128] @ w2[128,7->16], K=128 in 32 steps ----
  const int n2 = m;
  v8f acc2;
  {
    const float bb2 = (n2 < RDIR) ? b2[n2] : 0.f;
#pragma unroll
    for (int j = 0; j < 8; j++) acc2[j] = bb2;
  }
#pragma unroll 2
  for (int k = 0; k < 32; k++) {
    const int col = k * 4 + khalf;
    v2f a;
    a.x = hlds[m * HSTRIDE + col];
    a.y = hlds[m * HSTRIDE + col + 1];
    const v2f bf = *(const v2f*)(w2p + (k * 32 + lane) * 2);
    acc2 = __builtin_amdgcn_wmma_f32_16x16x4_f32(
        false, a, false, bf, (short)0, acc2, false, false);
  }

  // ---- Epilogue: occupancy (via lane shuffle) + maneuver mask, scatter [B,56] ----
  if (n2 < RDIR) {
#pragma unroll
    for (int j = 0; j < 8; j++) {
      const int M    = j + mrow;
      const int rowj = tile * 16 + M;
      const int bg   = rowj >> 3;
      const int g    = rowj & 7;
      const int oidx = bg * (GMAX * RDIR) + g * RDIR + n2;
      const bool occ = __shfl(idx, M, 32) >= 0;   // idx for row M lives in lane M
      const bool mv  = mvmask[oidx] != 0;
      out[oidx] = (occ && mv) ? acc2[j] : NEGV;
    }
  }
}

extern "C" void kernel_launch(void* const* d_in, const int* in_sizes, int n_in,
                              void* d_out, int out_size, void* d_ws, size_t ws_size,
                              hipStream_t stream) {
  const float* nf     = (const float*)d_in[0];
  const float* gf     = (const float*)d_in[1];
  const int*   gmask  = (const int*)d_in[2];
  const int*   mvmask = (const int*)d_in[3];
  const int*   batch  = (const int*)d_in[4];
  const float* w1     = (const float*)d_in[5];
  const float* b1     = (const float*)d_in[6];
  const float* w2     = (const float*)d_in[7];
  const float* b2     = (const float*)d_in[8];
  float* out = (float*)d_out;

  const int N = in_sizes[0] / DIM;   // 262144
  const int B = in_sizes[1] / DIM;   // 4096

  char* ws = (char*)d_ws;
  int*   idxbuf = (int*)ws;                                // B*8*4   = 128 KB
  float* zrow   = (float*)(ws + (size_t)B * GMAX * 4);     // 512 B
  float* w1p    = (float*)(ws + (size_t)B * GMAX * 4 + 512);           // 128 KB
  float* w2p    = (float*)(ws + (size_t)B * GMAX * 4 + 512 + 131072);  // 8 KB

  // Weight repack + zero row (tiny, runs concurrently-independent of slots).
  const int packThreads = W1P_FRAGS + W2P_FRAGS + DIM;     // 17536
  mh_pack_kernel<<<dim3((packThreads + 255) / 256), dim3(256), 0, stream>>>(
      w1, w2, w1p, w2p, zrow);

  mh_slots_kernel<<<dim3(B), dim3(32), 0, stream>>>(batch, gmask, idxbuf, N);

  const int tiles = (B * GMAX) / 16; // 2048
  mh_mlp_kernel<<<dim3(tiles), dim3(32), 0, stream>>>(
      nf, gf, idxbuf, mvmask, w1p, b1, w2p, b2, zrow, out);
}